// FeaturePropagation_1211180777513
// MI455X (gfx1250) — compile-verified
//
#include <hip/hip_runtime.h>
#include <cfloat>

typedef __attribute__((ext_vector_type(2))) float v2f;
typedef __attribute__((ext_vector_type(8))) float v8f;
typedef __attribute__((ext_vector_type(4))) unsigned int u32x4;
typedef __attribute__((ext_vector_type(8))) int i32x8;
typedef __attribute__((ext_vector_type(4))) int i32x4;

// Fixed problem shape (from reference setup_inputs)
constexpr int Bc   = 8;
constexpr int Np   = 8192;
constexpr int Sp   = 2048;
constexpr int C1   = 128;
constexpr int C2   = 256;
constexpr int CIN  = 384;    // concat order: [interp(C2), features1(C1)]
constexpr int CMID = 256;
constexpr int COUT = 256;
constexpr int MTOT = Bc * Np;  // 65536
constexpr float BN_EPS = 1e-5f;

__device__ __forceinline__ void top3_ins(float d, int s,
    float& d0, int& i0, float& d1, int& i1, float& d2, int& i2) {
  if (d < d2) {
    if (d < d1) {
      d2 = d1; i2 = i1;
      if (d < d0) { d1 = d0; i1 = i0; d0 = d; i0 = s; }
      else        { d1 = d;  i1 = s; }
    } else { d2 = d; i2 = s; }
  }
}

// ---------------- 3-NN interpolation + concat -> cat (MTOT x 384) ----------
__global__ __launch_bounds__(256) void interp_cat_kernel(
    const float* __restrict__ xyz1, const float* __restrict__ xyz2,
    const float* __restrict__ f1, const float* __restrict__ f2,
    float* __restrict__ cat)
{
  __shared__ float sx[Sp], sy[Sp], sz[Sp], sn[Sp];
  const int tid = threadIdx.x;
  const int b = blockIdx.x / (Np / 8);
  const int nbase = (blockIdx.x % (Np / 8)) * 8;

  for (int i = tid; i < Sp * 3; i += 256) {
    float v = xyz2[(size_t)b * Sp * 3 + i];
    int s = i / 3, c = i - 3 * s;
    if (c == 0) sx[s] = v;
    else if (c == 1) sy[s] = v;
    else sz[s] = v;
  }
  __syncthreads();
  for (int s = tid; s < Sp; s += 256)
    sn[s] = sx[s] * sx[s] + sy[s] * sy[s] + sz[s] * sz[s];
  __syncthreads();

  const int w = tid >> 5, lane = tid & 31;
  const int n = nbase + w;
  const size_t prow = ((size_t)b * Np + n) * 3;
  const float px = xyz1[prow], py = xyz1[prow + 1], pz = xyz1[prow + 2];
  const float pn = px * px + py * py + pz * pz;

  float d0 = FLT_MAX, d1 = FLT_MAX, d2 = FLT_MAX;
  int   i0 = 0, i1 = 0, i2 = 0;
  for (int s = lane; s < Sp; s += 32) {
    float d = fmaxf(pn + sn[s] - 2.f * (px * sx[s] + py * sy[s] + pz * sz[s]), 0.f);
    top3_ins(d, s, d0, i0, d1, i1, d2, i2);
  }
  for (int off = 16; off > 0; off >>= 1) {
    float e0 = __shfl_xor(d0, off), e1 = __shfl_xor(d1, off), e2 = __shfl_xor(d2, off);
    int   j0 = __shfl_xor(i0, off), j1 = __shfl_xor(i1, off), j2 = __shfl_xor(i2, off);
    top3_ins(e0, j0, d0, i0, d1, i1, d2, i2);
    top3_ins(e1, j1, d0, i0, d1, i1, d2, i2);
    top3_ins(e2, j2, d0, i0, d1, i1, d2, i2);
  }

  const float r0 = 1.f / (d0 + 1e-8f), r1 = 1.f / (d1 + 1e-8f), r2 = 1.f / (d2 + 1e-8f);
  const float rs = 1.f / (r0 + r1 + r2);
  const float w0 = r0 * rs, w1 = r1 * rs, w2 = r2 * rs;

  const float4* ra = (const float4*)(f2 + ((size_t)b * Sp + i0) * C2);
  const float4* rb = (const float4*)(f2 + ((size_t)b * Sp + i1) * C2);
  const float4* rc = (const float4*)(f2 + ((size_t)b * Sp + i2) * C2);
  float* crow = cat + ((size_t)b * Np + n) * CIN;

  #pragma unroll
  for (int q = 0; q < 2; ++q) {
    int c4 = lane * 2 + q;
    float4 a = ra[c4], bq = rb[c4], cq = rc[c4];
    float4 o;
    o.x = w0 * a.x + w1 * bq.x + w2 * cq.x;
    o.y = w0 * a.y + w1 * bq.y + w2 * cq.y;
    o.z = w0 * a.z + w1 * bq.z + w2 * cq.z;
    o.w = w0 * a.w + w1 * bq.w + w2 * cq.w;
    ((float4*)crow)[c4] = o;
  }
  float4 fv = ((const float4*)(f1 + ((size_t)b * Np + n) * C1))[lane];
  ((float4*)(crow + C2))[lane] = fv;
}

// ---------------- TDM 2-D tile load: global (row-major) -> padded LDS ------
// Packs a D# per ISA §8: count=1, type=2, data_size=4B, optional LDS padding.
__device__ __forceinline__ void tdm_load_2d(unsigned int lds_off, const void* gptr,
    int cols, int rows, int rowStrideElts, int padInterval, int padAmount)
{
  unsigned long long ga = (unsigned long long)(uintptr_t)gptr;
  u32x4 g0;
  g0[0] = 1u;                                   // count=1, no gather
  g0[1] = lds_off;                              // LDS byte address
  g0[2] = (unsigned int)ga;                     // global_addr[31:0]
  g0[3] = (unsigned int)((ga >> 32) & 0x1FFFFFFu) | (2u << 30);  // addr hi | type=2
  i32x8 g1;
  unsigned int d0 = (2u << 16)                  // data_size = 4 bytes
                  | (1u << 20)                  // pad_enable
                  | ((unsigned int)padInterval << 22)
                  | ((unsigned int)padAmount << 25);
  g1[0] = (int)d0;
  g1[1] = (int)(((unsigned int)cols & 0xFFFFu) << 16);                 // tensor_dim0 lo
  g1[2] = (int)((((unsigned int)cols >> 16) & 0xFFFFu)
              | (((unsigned int)rows & 0xFFFFu) << 16));               // dim0 hi | dim1 lo
  g1[3] = (int)((((unsigned int)rows >> 16) & 0xFFFFu)
              | (((unsigned int)cols & 0xFFFFu) << 16));               // dim1 hi | tile_dim0
  g1[4] = (int)((unsigned int)rows & 0xFFFFu);                         // tile_dim1 (tile_dim2=0)
  g1[5] = (int)(unsigned int)rowStrideElts;                            // tensor_dim0_stride lo32
  g1[6] = 0;
  g1[7] = 0;
  i32x4 z4 = {0, 0, 0, 0};
#if __has_include(<hip/amd_detail/amd_gfx1250_TDM.h>)
  i32x8 z8 = {};
  __builtin_amdgcn_tensor_load_to_lds(g0, g1, z4, z4, z8, 0);
#else
  __builtin_amdgcn_tensor_load_to_lds(g0, g1, z4, z4, 0);
#endif
}

// ---------------- WMMA FP32 GEMM with TDM double buffering -----------------
// C(MxCOUT) = A(MxK) * Bw(KxCOUT) + bias.  WG: 8 waves (2Mx4N), wave tile 32x32.
constexpr int KC = 32;       // K-chunk depth
constexpr int ASTR = 36;     // A LDS row stride (32 + 4 pad)
constexpr int BSTR = 136;    // B LDS row stride (128 + 8 pad)

__global__ __launch_bounds__(256) void gemm_wmma_tdm(
    const float* __restrict__ A, const float* __restrict__ Bw,
    const float* __restrict__ bias, float* __restrict__ C, int K)
{
  __shared__ float As[2][64][ASTR];   // 18432 B
  __shared__ float Bs[2][KC][BSTR];   // 34816 B
  const int tid  = threadIdx.x;
  const int n0   = blockIdx.x * 128;
  const int m0   = blockIdx.y * 64;
  const int lane = tid & 31;
  const int wv   = tid >> 5;
  const int mw   = wv & 1;     // 2 M-groups of 32
  const int nw   = wv >> 1;    // 4 N-groups of 32
  const int half = lane >> 4;
  const int lr   = lane & 15;
  const int wvS  = __builtin_amdgcn_readfirstlane(wv);  // scalar wave id

  const int arow0 = mw * 32 + lr;
  const int arow1 = arow0 + 16;
  const int bcol0 = nw * 32 + lr;
  const int bcol1 = bcol0 + 16;

  v8f acc00 = {}, acc01 = {}, acc10 = {}, acc11 = {};

  const int nChunks = K / KC;
  if (wvS == 0) {
    tdm_load_2d((unsigned int)(uintptr_t)&As[0][0][0],
                A + (size_t)m0 * K, KC, 64, K, /*interval 32dw*/4, /*pad 4dw*/3);
    tdm_load_2d((unsigned int)(uintptr_t)&Bs[0][0][0],
                Bw + (size_t)0 * COUT + n0, 128, KC, COUT, /*interval 128dw*/6, /*pad 8dw*/7);
  }

  for (int i = 0; i < nChunks; ++i) {
    const int cur = i & 1;
    if (wvS == 0)
      __builtin_amdgcn_s_wait_tensorcnt(0);   // current buffers landed in LDS
    __syncthreads();
    if (wvS == 0 && (i + 1) < nChunks) {      // prefetch next chunk into other buffer
      const int k0n = (i + 1) * KC;
      tdm_load_2d((unsigned int)(uintptr_t)&As[1 - cur][0][0],
                  A + (size_t)m0 * K + k0n, KC, 64, K, 4, 3);
      tdm_load_2d((unsigned int)(uintptr_t)&Bs[1 - cur][0][0],
                  Bw + (size_t)k0n * COUT + n0, 128, KC, COUT, 6, 7);
    }

    #pragma unroll
    for (int kk = 0; kk < KC; kk += 4) {
      v2f a0 = *(const v2f*)&As[cur][arow0][kk + 2 * half];
      v2f a1 = *(const v2f*)&As[cur][arow1][kk + 2 * half];
      v2f b0, b1;
      b0.x = Bs[cur][kk + 2 * half][bcol0];
      b0.y = Bs[cur][kk + 2 * half + 1][bcol0];
      b1.x = Bs[cur][kk + 2 * half][bcol1];
      b1.y = Bs[cur][kk + 2 * half + 1][bcol1];
      acc00 = __builtin_amdgcn_wmma_f32_16x16x4_f32(false, a0, false, b0, (short)0, acc00, false, false);
      acc01 = __builtin_amdgcn_wmma_f32_16x16x4_f32(false, a0, false, b1, (short)0, acc01, false, false);
      acc10 = __builtin_amdgcn_wmma_f32_16x16x4_f32(false, a1, false, b0, (short)0, acc10, false, false);
      acc11 = __builtin_amdgcn_wmma_f32_16x16x4_f32(false, a1, false, b1, (short)0, acc11, false, false);
    }
    __syncthreads();   // all reads of 'cur' done before it is refilled
  }

  const int col0 = n0 + nw * 32 + lr;
  const int col1 = col0 + 16;
  const float bv0 = bias[col0], bv1 = bias[col1];
  const int rb0 = m0 + mw * 32 + half * 8;
  const int rb1 = rb0 + 16;
  #pragma unroll
  for (int r = 0; r < 8; ++r) {
    C[(size_t)(rb0 + r) * COUT + col0] = acc00[r] + bv0;
    C[(size_t)(rb0 + r) * COUT + col1] = acc01[r] + bv1;
    C[(size_t)(rb1 + r) * COUT + col0] = acc10[r] + bv0;
    C[(size_t)(rb1 + r) * COUT + col1] = acc11[r] + bv1;
  }
}

// ---------------- BN batch statistics (column sum / sumsq) -----------------
__global__ __launch_bounds__(256) void bn_stats(const float* __restrict__ Y,
    float* __restrict__ sum, float* __restrict__ sumsq)
{
  const int c  = threadIdx.x;
  const int r0 = blockIdx.x * 256;
  float s = 0.f, s2 = 0.f;
  for (int r = 0; r < 256; ++r) {
    float v = Y[(size_t)(r0 + r) * COUT + c];
    s += v; s2 += v * v;
  }
  atomicAdd(&sum[c], s);
  atomicAdd(&sumsq[c], s2);
}

__global__ void bn_finalize(const float* __restrict__ sum,
    const float* __restrict__ sumsq, const float* __restrict__ g,
    const float* __restrict__ be, float* __restrict__ scale,
    float* __restrict__ shift)
{
  int c = threadIdx.x;
  float mean = sum[c] * (1.f / (float)MTOT);
  float var  = sumsq[c] * (1.f / (float)MTOT) - mean * mean;
  float s    = g[c] * rsqrtf(var + BN_EPS);
  scale[c] = s;
  shift[c] = be[c] - mean * s;
}

__global__ __launch_bounds__(256) void bn_apply_relu(const float* __restrict__ Y,
    const float* __restrict__ scale, const float* __restrict__ shift,
    float* __restrict__ out)
{
  size_t q = (size_t)blockIdx.x * 256 + threadIdx.x;  // float4 index
  int c4 = (int)(q & 63);
  float4 v  = ((const float4*)Y)[q];
  float4 sc = ((const float4*)scale)[c4];
  float4 sh = ((const float4*)shift)[c4];
  float4 o;
  o.x = fmaxf(0.f, v.x * sc.x + sh.x);
  o.y = fmaxf(0.f, v.y * sc.y + sh.y);
  o.z = fmaxf(0.f, v.z * sc.z + sh.z);
  o.w = fmaxf(0.f, v.w * sc.w + sh.w);
  ((float4*)out)[q] = o;
}

__global__ void zero_stats(float* __restrict__ p, int n) {
  int i = blockIdx.x * blockDim.x + threadIdx.x;
  if (i < n) p[i] = 0.f;
}

extern "C" void kernel_launch(void* const* d_in, const int* in_sizes, int n_in,
                              void* d_out, int out_size, void* d_ws, size_t ws_size,
                              hipStream_t stream) {
  (void)in_sizes; (void)n_in; (void)out_size; (void)ws_size;
  const float* xyz1 = (const float*)d_in[0];
  const float* xyz2 = (const float*)d_in[1];
  const float* f1   = (const float*)d_in[2];
  const float* f2   = (const float*)d_in[3];
  const float* W1   = (const float*)d_in[4];
  const float* b1   = (const float*)d_in[5];
  const float* g1   = (const float*)d_in[6];
  const float* be1  = (const float*)d_in[7];
  const float* W2   = (const float*)d_in[8];
  const float* b2   = (const float*)d_in[9];
  const float* g2   = (const float*)d_in[10];
  const float* be2  = (const float*)d_in[11];
  float* out = (float*)d_out;

  // workspace layout (floats): cat | y1 | stats ; y2 overlays cat
  float* ws   = (float*)d_ws;
  float* cat  = ws;                                   // MTOT*CIN
  float* y1   = ws + (size_t)MTOT * CIN;              // MTOT*CMID
  float* st   = y1 + (size_t)MTOT * CMID;             // 2048 floats of stats
  float* sum1 = st,        *sq1 = st + 256,  *scale1 = st + 512,  *shift1 = st + 768;
  float* sum2 = st + 1024, *sq2 = st + 1280, *scale2 = st + 1536, *shift2 = st + 1792;
  float* y2   = cat;  // cat dead after GEMM1 consumes it

  zero_stats<<<8, 256, 0, stream>>>(st, 2048);
  interp_cat_kernel<<<Bc * Np / 8, 256, 0, stream>>>(xyz1, xyz2, f1, f2, cat);

  gemm_wmma_tdm<<<dim3(COUT / 128, MTOT / 64), 256, 0, stream>>>(cat, W1, b1, y1, CIN);
  bn_stats<<<MTOT / 256, 256, 0, stream>>>(y1, sum1, sq1);
  bn_finalize<<<1, 256, 0, stream>>>(sum1, sq1, g1, be1, scale1, shift1);
  bn_apply_relu<<<(MTOT * CMID / 4) / 256, 256, 0, stream>>>(y1, scale1, shift1, y1);

  gemm_wmma_tdm<<<dim3(COUT / 128, MTOT / 64), 256, 0, stream>>>(y1, W2, b2, y2, CMID);
  bn_stats<<<MTOT / 256, 256, 0, stream>>>(y2, sum2, sq2);
  bn_finalize<<<1, 256, 0, stream>>>(sum2, sq2, g2, be2, scale2, shift2);
  bn_apply_relu<<<(MTOT * COUT / 4) / 256, 256, 0, stream>>>(y2, scale2, shift2, out);
}